// Black_oil_peacemann_13116830122018
// MI455X (gfx1250) — compile-verified
//
#include <hip/hip_runtime.h>
#include <math.h>

typedef __attribute__((ext_vector_type(2))) float v2f;
typedef __attribute__((ext_vector_type(4))) float v4f;
typedef __attribute__((ext_vector_type(8))) float v8f;

#define NBATCH 4096
#define FX 89
#define FY 66
#define W  128
#define WV (W / 4)          // 32 float4 per row
#define XSTR (FX * W)
#define YSTR (FY * W)

// ---------------------------------------------------------------------------
// Kernel 1: per-batch p_mean via V_WMMA_F32_16X16X4_F32, then the three
// per-batch Peaceman coefficients written to workspace.
//   ws[0*NBATCH + n] = C_oil, ws[1*NBATCH + n] = C_water, ws[2*NBATCH + n] = C_gas
// One wave (32 threads) handles 16 batches. Row-sum of P(16x128) is computed
// as D = sum_t A_t(16x4) x Ones(4x16): every column of D carries the row sum.
// ---------------------------------------------------------------------------
__global__ __launch_bounds__(32) void pmean_coeff_kernel(
    const float* __restrict__ X, float* __restrict__ ws) {
  __shared__ float pbuf[16 * W];
  __shared__ float pm[16];

  const int lane = threadIdx.x;
  const int n0   = blockIdx.x * 16;

  // Stage 16 pressure rows (X[n, 22, :]) into LDS, coalesced B128 along j.
  {
    v4f* pb4 = (v4f*)pbuf;
    for (int i = lane; i < 16 * WV; i += 32) {
      const int b  = i >> 5;        // batch within block
      const int j4 = i & (WV - 1);  // float4 column
      const v4f* src =
          (const v4f*)(X + (size_t)(n0 + b) * XSTR + 22 * W) + j4;
      pb4[i] = *src;
    }
  }
  __syncthreads();

  // A-fragment layout (32-bit A, 16x4): lanes 0-15 hold K=0,1 in v0,v1;
  // lanes 16-31 hold K=2,3.  m = lane&15, koff = (lane>=16) ? 2 : 0.
  const int m    = lane & 15;
  const int koff = (lane >> 4) << 1;

  v2f onesB;
  onesB.x = 1.0f;
  onesB.y = 1.0f;
  v8f acc = {};

  for (int t = 0; t < 32; ++t) {
    const int kb = 4 * t + koff;
    v2f a;
    a.x = pbuf[m * W + kb];
    a.y = pbuf[m * W + kb + 1];
    acc = __builtin_amdgcn_wmma_f32_16x16x4_f32(
        /*neg_a=*/false, a, /*neg_b=*/false, onesB,
        /*c_mod=*/(short)0, acc, /*reuse_a=*/false, /*reuse_b=*/false);
  }

  // C/D layout: VGPR r holds row M=r (lanes 0-15) and M=8+r (lanes 16-31);
  // with all-ones B every column equals the row sum.
  if (lane == 0) {
#pragma unroll
    for (int r = 0; r < 8; ++r) pm[r] = acc[r] * (1.0f / 128.0f);
  }
  if (lane == 16) {
#pragma unroll
    for (int r = 0; r < 8; ++r) pm[8 + r] = acc[r] * (1.0f / 128.0f);
  }
  __syncthreads();

  if (lane < 16) {
    const float p = pm[lane];

    // calc_bg / calc_bo / calc_mu_g on the per-batch scalar (fp32, as ref).
    const float P_BUB = 0.5f, P_ATM = 0.1f;
    const float dp = (p < P_BUB) ? (P_ATM - p) : (P_ATM - P_BUB);
    const float bg = 1.0f / expf(0.0017f * dp);

    const float e1 = (p < P_BUB) ? (-8e-5f * (P_ATM - p))
                                 : (-8e-5f * (P_ATM - P_BUB));
    const float e2 = -1e-5f * ((p < P_BUB) ? 0.0f : (p - P_BUB));
    const float bo = 1.0f / (expf(e1) * expf(e2));

    const float mu_g = 3e-10f * p * p + 1e-6f * p + 0.0133f;

    // right = log(RE/RWELL)+SKIN = ln(2);  K = 2*pi*DZ/right
    const float right = 0.6931471805599453f;
    const float K     = 6.283185307179586f * 100.0f / right;
    const float add   = fabsf(p - 100.0f);  // |drawdown|; kr,perm >= 0

    const int n = n0 + lane;
    ws[0 * NBATCH + n] = add * K / (2.5f * bo);     // oil:   UO * bo
    ws[1 * NBATCH + n] = add * K;                   // water: UW * BW = 1
    ws[2 * NBATCH + n] = add * K / (mu_g * bg);     // gas:   mu_g * bg
  }
}

// ---------------------------------------------------------------------------
// Kernel 2: streaming elementwise pass. One block per batch (8 waves);
// wave w owns rows k = w, w+8, w+16; lane owns a column quad. All traffic is
// B128 coalesced (512B/wave) with non-temporal hints (pure one-shot stream).
// ---------------------------------------------------------------------------
__global__ __launch_bounds__(256) void loss_kernel(
    const float* __restrict__ X, const float* __restrict__ Y,
    const float* __restrict__ ws, float* __restrict__ out) {
  const int n = blockIdx.x;
  const int c = threadIdx.x & 31;   // float4 column index (j = 4c)
  const int w = threadIdx.x >> 5;   // wave id 0..7

  const v4f* Xv = (const v4f*)(X + (size_t)n * XSTR);
  const v4f* Yv = (const v4f*)(Y + (size_t)n * YSTR);
  v4f*       Ov = (v4f*)(out + (size_t)n * YSTR);

  const float co = ws[0 * NBATCH + n];
  const float cw = ws[1 * NBATCH + n];
  const float cg = ws[2 * NBATCH + n];

  const float S    = 1e-10f / 4096.0f;  // *1e-10 and /N folded together
  const float invD = 1.0f / 0.7f;       // 1/(1-SWI-SOR)
  const float SWI  = 0.1f;

  for (int k = w; k < 22; k += 8) {
    const v4f perm = __builtin_nontemporal_load(&Xv[k * WV + c]);
    const v4f sg   = __builtin_nontemporal_load(&Xv[(45 + k) * WV + c]);
    const v4f sw   = __builtin_nontemporal_load(&Xv[(67 + k) * WV + c]);
    const v4f y0   = __builtin_nontemporal_load(&Yv[k * WV + c]);
    const v4f y1   = __builtin_nontemporal_load(&Yv[(22 + k) * WV + c]);
    const v4f y2   = __builtin_nontemporal_load(&Yv[(44 + k) * WV + c]);

    const v4f swp = (sw - SWI) * invD;
    const v4f sgp = sg * invD;
    const v4f osw = 1.0f - swp;
    const v4f osg = 1.0f - sgp;

    const v4f krw = 0.3f * swp * swp;
    const v4f kro = 0.9f * (osw * osw) * (osg * osg);
    const v4f krg = 0.8f * sgp * sgp;

    const v4f r0 = (co * perm * kro - y0) * S;
    const v4f r1 = (cw * perm * krw - y1) * S;
    const v4f r2 = (cg * perm * krg - y2) * S;

    __builtin_nontemporal_store(r0, &Ov[k * WV + c]);
    __builtin_nontemporal_store(r1, &Ov[(22 + k) * WV + c]);
    __builtin_nontemporal_store(r2, &Ov[(44 + k) * WV + c]);
  }
}

extern "C" void kernel_launch(void* const* d_in, const int* in_sizes, int n_in,
                              void* d_out, int out_size, void* d_ws, size_t ws_size,
                              hipStream_t stream) {
  (void)in_sizes; (void)n_in; (void)out_size; (void)ws_size;
  const float* X = (const float*)d_in[0];   // (4096, 89, 128) fp32
  const float* Y = (const float*)d_in[1];   // (4096, 66, 128) fp32
  float* out = (float*)d_out;               // (4096, 66, 128) fp32
  float* ws  = (float*)d_ws;                // needs 3*4096*4 = 48 KB

  pmean_coeff_kernel<<<NBATCH / 16, 32, 0, stream>>>(X, ws);
  loss_kernel<<<NBATCH, 256, 0, stream>>>(X, Y, ws, out);
}